// Model_39676907884531
// MI455X (gfx1250) — compile-verified
//
#include <hip/hip_runtime.h>

typedef __attribute__((ext_vector_type(2))) float v2f;
typedef __attribute__((ext_vector_type(8))) float v8f;

#define H_IN  224
#define W_IN_ 224
#define HOUT  222
#define WOUT  222
#define COUT_ 288
#define BATCH 16
#define HW    (HOUT * WOUT)

// One wave computes a 16(channel) x 16(ow) output tile at a fixed oh, for all
// 16 batch images. GEMM view: D[16,16] = A[16,K=12] * B[12,16], K padded from
// the 9 conv taps; done as 3 chained V_WMMA_F32_16X16X4_F32.
__global__ __launch_bounds__(32) void conv3x3_relu_wmma(
    const float* __restrict__ x,     // (16,1,224,224)
    const float* __restrict__ w,     // (288,1,3,3)
    const float* __restrict__ bias,  // (1,288,222,222)
    float* __restrict__ out)         // (16,288,222,222)
{
    const int lane = threadIdx.x;          // 0..31
    const int n    = lane & 15;            // column within tile
    const bool hi  = lane >= 16;

    const int ow0 = blockIdx.x * 16;
    const int oh  = blockIdx.y;
    const int co0 = blockIdx.z * 16;

    const int nvalid = (WOUT - ow0) < 16 ? (WOUT - ow0) : 16;

    // ---- A fragments: weights, 16x4 f32 layout ----
    // lane<16: (M=lane) holds K0 in .x, K1 in .y ; lane>=16: (M=lane-16) K2,K3
    const float* wb = w + (co0 + n) * 9;
    v2f a0, a1, a2;
    if (!hi) {
        a0.x = wb[0]; a0.y = wb[1];     // taps 0,1
        a1.x = wb[4]; a1.y = wb[5];     // taps 4,5
        a2.x = wb[8]; a2.y = 0.0f;      // tap 8, pad
    } else {
        a0.x = wb[2]; a0.y = wb[3];     // taps 2,3
        a1.x = wb[6]; a1.y = wb[7];     // taps 6,7
        a2.x = 0.0f;  a2.y = 0.0f;      // pad
    }

    // ---- C/D layout: VGPR v, lane<16 -> (M=v, N=lane); lane>=16 -> (M=v+8) ----
    const int co_d = co0 + (hi ? 8 : 0);
    const int n_ld = (n < nvalid) ? n : (nvalid - 1);   // clamp for safe loads
    const int ow_st = ow0 + n;

    // Bias loaded ONCE per tile; reused across all 16 batches (bias is
    // batch-broadcast, so HBM bias traffic drops 16x and stays L2-resident).
    float bv[8];
    {
        const float* bp = bias + (size_t)co_d * HW + (size_t)oh * WOUT + (ow0 + n_ld);
        #pragma unroll
        for (int v = 0; v < 8; ++v) bv[v] = bp[(size_t)v * HW];
    }

    // ---- Branchless per-lane element offsets for the B fragments ----
    // Clamped input columns (last ow-tile may probe 2 past the row end)
    const int c0 = (ow0 + n + 0 < W_IN_) ? (ow0 + n + 0) : (W_IN_ - 1);
    const int c1 = (ow0 + n + 1 < W_IN_) ? (ow0 + n + 1) : (W_IN_ - 1);
    const int c2 = (ow0 + n + 2 < W_IN_) ? (ow0 + n + 2) : (W_IN_ - 1);

    // B layout 4x16 f32: VGPR0 = rows K0 (lanes<16) | K2 (lanes>=16),
    //                    VGPR1 = rows K1 | K3 ; tap r = ky*3+kx.
    const int rbase = oh * W_IN_;
    const int off_b0x = rbase + (hi ? (0 * W_IN_ + c2) : (0 * W_IN_ + c0)); // tap2 | tap0
    const int off_b0y = rbase + (hi ? (1 * W_IN_ + c0) : (0 * W_IN_ + c1)); // tap3 | tap1
    const int off_b1x = rbase + (hi ? (2 * W_IN_ + c0) : (1 * W_IN_ + c1)); // tap6 | tap4
    const int off_b1y = rbase + (hi ? (2 * W_IN_ + c1) : (1 * W_IN_ + c2)); // tap7 | tap5
    const int off_b2x = rbase + (2 * W_IN_ + c2);                           // tap8 (zeroed on hi)

    for (int b = 0; b < BATCH; ++b) {
        const float* xb = x + (size_t)b * (H_IN * W_IN_);

        v2f b0, b1, b2;
        b0.x = xb[off_b0x];
        b0.y = xb[off_b0y];
        b1.x = xb[off_b1x];
        b1.y = xb[off_b1y];
        float t = xb[off_b2x];
        b2.x = hi ? 0.0f : t;
        b2.y = 0.0f;

        v8f acc = {};
        acc = __builtin_amdgcn_wmma_f32_16x16x4_f32(false, a0, false, b0,
                                                    (short)0, acc, false, false);
        acc = __builtin_amdgcn_wmma_f32_16x16x4_f32(false, a1, false, b1,
                                                    (short)0, acc, false, false);
        acc = __builtin_amdgcn_wmma_f32_16x16x4_f32(false, a2, false, b2,
                                                    (short)0, acc, false, false);

        // ---- Epilogue: bias + ReLU + masked, coalesced, NON-TEMPORAL stores.
        // Output is write-once streaming data (908 MB): keep it from evicting
        // the reused bias/x working set in L2.
        if (n < nvalid) {
            float* op = out + (((size_t)b * COUT_ + co_d) * HOUT + oh) * WOUT + ow_st;
            #pragma unroll
            for (int v = 0; v < 8; ++v) {
                float o = acc[v] + bv[v];
                o = (o > 0.0f) ? o : 0.0f;
                __builtin_nontemporal_store(o, op + (size_t)v * HW);
            }
        }
    }
}

extern "C" void kernel_launch(void* const* d_in, const int* in_sizes, int n_in,
                              void* d_out, int out_size, void* d_ws, size_t ws_size,
                              hipStream_t stream) {
    const float* x    = (const float*)d_in[0];
    const float* w    = (const float*)d_in[1];
    const float* bias = (const float*)d_in[2];
    float* out = (float*)d_out;

    dim3 grid((WOUT + 15) / 16, HOUT, COUT_ / 16);   // (14, 222, 18)
    conv3x3_relu_wmma<<<grid, 32, 0, stream>>>(x, w, bias, out);
}